// L2_ChamferLoss_45337674776760
// MI455X (gfx1250) — compile-verified
//
#include <hip/hip_runtime.h>

// CDNA5 / gfx1250 Chamfer loss.
// Core op: V_WMMA_F32_16X16X4_F32 computing full 16x16 squared-distance tiles:
//   A = [-2ax,-2ay,-2az, 1]  (16x4),  B = [bx,by,bz,|b|^2] (4x16),  C = |a|^2 per row
//   D = A*B + C = |a|^2 + |b|^2 - 2 a.b
// b-points are staged in LDS pre-swizzled into the exact WMMA B-fragment layout
// (per tile: slot[lane<16] = (x,y) = K{0,1}, slot[lane>=16] = (z,|b|^2) = K{2,3})
// so the hot loop is just: ds_load_b64 -> wmma -> min3 tree -> ds_min_u32.
// Mins are taken unclamped in-loop (max(.,0) commutes with min); clamp applied
// once per tile (colmin) and once per wave (rowmin).

typedef __attribute__((ext_vector_type(2))) float v2f;
typedef __attribute__((ext_vector_type(8))) float v8f;

#define BATCH          8
#define NPTS           4096
#define MCHUNK         1024
#define TILES          (MCHUNK / 16)     // 64 tiles per chunk
#define ROWS_PER_BLOCK 128
#define THREADS        256   // 8 wave32

__global__ __launch_bounds__(THREADS)
void chamfer_init_kernel(unsigned int* __restrict__ dist2, int n) {
    int i = blockIdx.x * blockDim.x + threadIdx.x;
    if (i < n) dist2[i] = 0x7F800000u;   // +inf bit pattern
}

__global__ __launch_bounds__(THREADS)
void chamfer_wmma_kernel(const float* __restrict__ A,
                         const float* __restrict__ Bp,
                         float* __restrict__ dist1,
                         unsigned int* __restrict__ dist2) {
    // B fragments in WMMA layout, +32 pad slots so the pipelined prefetch of
    // tile (mt+1) is branch-free on the last iteration.
    __shared__ float2       bfrag_lds[MCHUNK * 2 + 32];   // ~16.25 KB
    __shared__ unsigned int colmin[MCHUNK];               // 4 KB
    __shared__ float        a2s[ROWS_PER_BLOCK];          // 512 B

    const int tid  = threadIdx.x;
    const int lane = tid & 31;
    const int wave = tid >> 5;
    const int blocksPerBatch = NPTS / ROWS_PER_BLOCK;   // 32
    const int batch    = blockIdx.x / blocksPerBatch;
    const int rowBase  = (blockIdx.x % blocksPerBatch) * ROWS_PER_BLOCK;

    const float* Ab = A  + (size_t)batch * NPTS * 3;
    const float* Bb = Bp + (size_t)batch * NPTS * 3;

    // |a|^2 for this block's 128 rows
    if (tid < ROWS_PER_BLOCK) {
        int r = rowBase + tid;
        float x = Ab[3 * r + 0], y = Ab[3 * r + 1], z = Ab[3 * r + 2];
        a2s[tid] = x * x + y * y + z * z;
    }

    // ---- A fragment (16x4 f32): lanes 0-15 hold K={0,1}, lanes 16-31 K={2,3}
    const int waveRow = rowBase + wave * 16;          // this wave's 16 rows
    const int hi = (lane >= 16) ? 8 : 0;
    v2f afrag;
    {
        int r = waveRow + (lane & 15);
        float x = Ab[3 * r + 0], y = Ab[3 * r + 1], z = Ab[3 * r + 2];
        afrag.x = (lane < 16) ? (-2.0f * x) : (-2.0f * z);
        afrag.y = (lane < 16) ? (-2.0f * y) : 1.0f;
    }

    __syncthreads();

    // ---- C accumulator: |a|^2 broadcast along each output row
    // C/D layout: VGPR v, lanes 0-15 -> M=v, lanes 16-31 -> M=v+8
    v8f cinit;
#pragma unroll
    for (int v = 0; v < 8; ++v) cinit[v] = a2s[wave * 16 + v + hi];

    float rmin[8];
#pragma unroll
    for (int v = 0; v < 8; ++v) rmin[v] = INFINITY;

    const int colBase = lane & 15;

    for (int ch = 0; ch < NPTS / MCHUNK; ++ch) {
        __syncthreads();   // previous chunk fully consumed before overwrite
        for (int j = tid; j < MCHUNK; j += THREADS) {
            int m = ch * MCHUNK + j;
            float x = Bb[3 * m + 0], y = Bb[3 * m + 1], z = Bb[3 * m + 2];
            int t = j >> 4, i = j & 15;
            bfrag_lds[t * 32 + i]      = make_float2(x, y);                    // K0,K1
            bfrag_lds[t * 32 + 16 + i] = make_float2(z, x * x + y * y + z * z); // K2,K3
            colmin[j] = 0x7F800000u;
        }
        __syncthreads();

        // Software-pipelined tile loop: prefetch fragment for tile mt+1.
        float2 pc = bfrag_lds[lane];
#pragma unroll 4
        for (int mt = 0; mt < TILES; ++mt) {
            float2 pn = bfrag_lds[(mt + 1) * 32 + lane];   // pad makes this safe

            v2f bfrag;
            bfrag.x = pc.x;
            bfrag.y = pc.y;

            // One WMMA = finished 16x16 distance tile.
            v8f d = __builtin_amdgcn_wmma_f32_16x16x4_f32(
                /*neg_a=*/false, afrag, /*neg_b=*/false, bfrag,
                /*c_mod=*/(short)0, cinit, /*reuse_a=*/false, /*reuse_b=*/false);

            // Row mins: unclamped float domain (clamped once at the end).
#pragma unroll
            for (int v = 0; v < 8; ++v) rmin[v] = fminf(rmin[v], d[v]);

            // Column min over this lane's 8 rows, single clamp, uint-ordered atomic.
            float c01 = fminf(d[0], d[1]);
            float c23 = fminf(d[2], d[3]);
            float c45 = fminf(d[4], d[5]);
            float c67 = fminf(d[6], d[7]);
            float cmin = fminf(fminf(c01, c23), fminf(c45, c67));
            cmin = fmaxf(cmin, 0.0f);
            atomicMin(&colmin[mt * 16 + colBase], __float_as_uint(cmin));

            pc = pn;
        }
        __syncthreads();

        for (int j = tid; j < MCHUNK; j += THREADS)
            atomicMin(&dist2[(size_t)batch * NPTS + ch * MCHUNK + j], colmin[j]);
    }

    // ---- finish dist1: reduce each row across its 16-lane half
#pragma unroll
    for (int v = 0; v < 8; ++v) {
#pragma unroll
        for (int m = 1; m < 16; m <<= 1)
            rmin[v] = fminf(rmin[v], __shfl_xor(rmin[v], m, 32));
    }
    if ((lane & 15) == 0) {
#pragma unroll
        for (int v = 0; v < 8; ++v)
            dist1[(size_t)batch * NPTS + waveRow + hi + v] = fmaxf(rmin[v], 0.0f);
    }
}

__global__ __launch_bounds__(THREADS)
void chamfer_reduce_kernel(const float* __restrict__ dist1,
                           const unsigned int* __restrict__ dist2,
                           float* __restrict__ out) {
    __shared__ float sdata[THREADS];
    const int tid = threadIdx.x;
    const int n = BATCH * NPTS;
    float s = 0.0f;
    for (int i = tid; i < n; i += THREADS) s += dist1[i];
    for (int i = tid; i < n; i += THREADS) s += __uint_as_float(dist2[i]);
    sdata[tid] = s;
    __syncthreads();
    for (int st = THREADS / 2; st > 0; st >>= 1) {
        if (tid < st) sdata[tid] += sdata[tid + st];
        __syncthreads();
    }
    // N == M, so mean(dist1)+mean(dist2) = (sum1+sum2)/(B*N)
    if (tid == 0) out[0] = sdata[0] / (float)n;
}

extern "C" void kernel_launch(void* const* d_in, const int* in_sizes, int n_in,
                              void* d_out, int out_size, void* d_ws, size_t ws_size,
                              hipStream_t stream) {
    const float* a = (const float*)d_in[0];   // [8, 4096, 3] fp32
    const float* b = (const float*)d_in[1];   // [8, 4096, 3] fp32

    float*        dist1 = (float*)d_ws;                                   // 128 KB
    unsigned int* dist2 = (unsigned int*)((char*)d_ws +
                                          (size_t)BATCH * NPTS * sizeof(float)); // 128 KB

    const int n = BATCH * NPTS;
    chamfer_init_kernel<<<(n + THREADS - 1) / THREADS, THREADS, 0, stream>>>(dist2, n);
    chamfer_wmma_kernel<<<BATCH * (NPTS / ROWS_PER_BLOCK), THREADS, 0, stream>>>(
        a, b, dist1, dist2);
    chamfer_reduce_kernel<<<1, THREADS, 0, stream>>>(dist1, dist2, (float*)d_out);
}